// AggregateLocalEdgesAttention_24953759989856
// MI455X (gfx1250) — compile-verified
//
#include <hip/hip_runtime.h>
#include <stdint.h>

// ---------------------------------------------------------------------------
// AggregateLocalEdgesAttention (GAT-style scatter softmax + weighted scatter)
//   idx   = edge_index[0]                      (M)
//   denom = segment_sum(exp(att), idx, N)      (N)
//   out   = segment_sum(edges * exp(att)/denom[idx], idx, N)   (N, F=64)
// Memory-bound: ~430 MB traffic -> stream edges via CDNA5 async global->LDS
// DMA (ASYNCcnt) with double buffering; scatter with native f32 L2 atomics.
// ---------------------------------------------------------------------------

#define F            64
#define TPB          256
#define TILE_EDGES   64
#define TILE_FLOATS  (TILE_EDGES * F)   // 4096 floats = 16 KB per buffer

#if __has_builtin(__builtin_amdgcn_global_load_async_to_lds_b128)
#define HAVE_ASYNC_LDS 1
#else
#define HAVE_ASYNC_LDS 0
#endif

#if HAVE_ASYNC_LDS
#if __has_builtin(__builtin_amdgcn_s_wait_asynccnt)
#define WAIT_ASYNC(n) __builtin_amdgcn_s_wait_asynccnt(n)
#else
#define WAIT_ASYNC(n) asm volatile("s_wait_asynccnt %0" :: "i"(n) : "memory")
#endif
#else
#define WAIT_ASYNC(n) do {} while (0)
#endif

// Exact parameter types for the async-LDS builtin:
//   (AS1 v4i* global src, AS3 v4i* lds dst, imm offset, imm cpol)
typedef __attribute__((__vector_size__(4 * sizeof(int)))) int v4i;
typedef __attribute__((address_space(1))) v4i gv4i;
typedef __attribute__((address_space(3))) v4i lv4i;

// Relaxed agent-scope fp32 atomic add -> global_atomic_add_f32 (no return).
__device__ __forceinline__ void atomAddF(float* p, float v) {
    __hip_atomic_fetch_add(p, v, __ATOMIC_RELAXED, __HIP_MEMORY_SCOPE_AGENT);
}

// Stage one full tile (TILE_FLOATS floats) global -> LDS.
// 256 threads x 4 issues x 16B = 16 KB; 4 async instructions per wave.
__device__ __forceinline__ void issue_tile(const float* __restrict__ gsrc,
                                           float* ldst, int tid) {
#if HAVE_ASYNC_LDS
#pragma unroll
    for (int i = 0; i < 4; ++i) {
        const int off = (tid + i * TPB) * 4;  // float4 granularity
        gv4i* g = (gv4i*)(uintptr_t)(gsrc + off);
        // generic LDS pointer low 32 bits == LDS byte offset (aperture rule)
        lv4i* l = (lv4i*)(uintptr_t)(uint32_t)(uintptr_t)(ldst + off);
        __builtin_amdgcn_global_load_async_to_lds_b128(g, l, 0, 0);
    }
#else
#pragma unroll
    for (int i = 0; i < 4; ++i) {
        const int off = (tid + i * TPB) * 4;
        *(float4*)(ldst + off) = *(const float4*)(gsrc + off);
    }
#endif
}

__global__ void zero_k(float* __restrict__ p, int n) {
    const int i4 = (blockIdx.x * blockDim.x + threadIdx.x) * 4;
    if (i4 + 4 <= n) {
        *(float4*)(p + i4) = make_float4(0.f, 0.f, 0.f, 0.f);
    } else {
        for (int k = i4; k < n; ++k) p[k] = 0.f;
    }
}

__global__ void denom_k(const float* __restrict__ att,
                        const int* __restrict__ recv,
                        float* __restrict__ denom, int M) {
    const int e = blockIdx.x * blockDim.x + threadIdx.x;
    if (e < M) atomAddF(&denom[recv[e]], __expf(att[e]));
}

__global__ __launch_bounds__(TPB)
void scatter_k(const float* __restrict__ edges,
               const float* __restrict__ att,
               const int* __restrict__ recv,
               const float* __restrict__ denom,
               float* __restrict__ out,
               int M, int tilesTotal) {
    __shared__ float tile[2][TILE_FLOATS];
    __shared__ float s_alpha[2][TILE_EDGES];
    __shared__ int   s_idx[2][TILE_EDGES];

    const int tid    = threadIdx.x;
    const int stride = gridDim.x;
    int t = blockIdx.x;
    if (t >= tilesTotal) return;

    int  cur     = 0;
    bool curFull = ((long)(t + 1) * TILE_EDGES <= (long)M);
    if (curFull) issue_tile(edges + (long)t * TILE_FLOATS, tile[0], tid);

    for (; t < tilesTotal; t += stride) {
        const int  tn       = t + stride;
        const bool nextFull = (tn < tilesTotal) &&
                              ((long)(tn + 1) * TILE_EDGES <= (long)M);

        // Per-edge softmax weight + receiver id for this tile (L2-hot gathers).
        if (tid < TILE_EDGES) {
            const long e = (long)t * TILE_EDGES + tid;
            float a = 0.f;
            int   n = 0;
            if (e < (long)M) {
                n = recv[e];
                a = __expf(att[e]) / denom[n];
            }
            s_alpha[cur][tid] = a;
            s_idx[cur][tid]   = n;
        }

        // Kick off next tile's DMA, then drain everything except it.
        if (nextFull) {
            issue_tile(edges + (long)tn * TILE_FLOATS, tile[cur ^ 1], tid);
            WAIT_ASYNC(4);   // leave only the 4 just-issued per-wave loads
        } else {
            WAIT_ASYNC(0);
        }
        __syncthreads();      // tile[cur] + s_alpha/s_idx[cur] visible to all

        // 4 threads per edge, 16 floats each: scale + scatter-add.
        const int eloc = tid >> 2;
        const int part = (tid & 3) * 16;
        if (curFull) {
            const float a  = s_alpha[cur][eloc];
            const int   ni = s_idx[cur][eloc];
            float*       op = out + (long)ni * F + part;
            const float* ip = &tile[cur][eloc * F + part];
#pragma unroll
            for (int j = 0; j < 4; ++j) {
                const float4 v = *(const float4*)(ip + j * 4);
                atomAddF(op + j * 4 + 0, v.x * a);
                atomAddF(op + j * 4 + 1, v.y * a);
                atomAddF(op + j * 4 + 2, v.z * a);
                atomAddF(op + j * 4 + 3, v.w * a);
            }
        } else {
            const long e = (long)t * TILE_EDGES + eloc;
            if (e < (long)M) {
                const float a  = s_alpha[cur][eloc];
                const int   ni = s_idx[cur][eloc];
                float*       op = out + (long)ni * F + part;
                const float* ip = edges + e * F + part;
#pragma unroll
                for (int j = 0; j < 4; ++j) {
                    const float4 v = *(const float4*)(ip + j * 4);
                    atomAddF(op + j * 4 + 0, v.x * a);
                    atomAddF(op + j * 4 + 1, v.y * a);
                    atomAddF(op + j * 4 + 2, v.z * a);
                    atomAddF(op + j * 4 + 3, v.w * a);
                }
            }
        }
        __syncthreads();      // done reading tile[cur] before it is re-DMA'd
        cur ^= 1;
        curFull = nextFull;
    }
}

extern "C" void kernel_launch(void* const* d_in, const int* in_sizes, int n_in,
                              void* d_out, int out_size, void* d_ws, size_t ws_size,
                              hipStream_t stream) {
    // Input order: reference (N*F, unused), edges (M*F), attention (M),
    //              edge_index (2*M, row 0 = receiver ids)
    const float* edges = (const float*)d_in[1];
    const float* att   = (const float*)d_in[2];
    const int*   recv  = (const int*)d_in[3];   // first M entries = edge_index[0]

    const int NF = in_sizes[0];     // N * 64
    const int M  = in_sizes[2];     // number of edges
    const int N  = NF / F;

    float* out   = (float*)d_out;
    float* denom = (float*)d_ws;    // N floats of scratch

    // Zero output and denom scratch.
    {
        int nt = (NF + 3) / 4;
        zero_k<<<(nt + TPB - 1) / TPB, TPB, 0, stream>>>(out, NF);
        nt = (N + 3) / 4;
        zero_k<<<(nt + TPB - 1) / TPB, TPB, 0, stream>>>(denom, N);
    }

    // Pass 1: denom[n] = sum exp(att) over edges targeting n.
    denom_k<<<(M + TPB - 1) / TPB, TPB, 0, stream>>>(att, recv, denom, M);

    // Pass 2: weighted scatter with async-LDS streaming, persistent blocks.
    const int tiles = (M + TILE_EDGES - 1) / TILE_EDGES;
    const int grid  = tiles < 2048 ? tiles : 2048;
    scatter_k<<<grid, TPB, 0, stream>>>(edges, att, recv, denom, out, M, tiles);
}